// ReadUnit_41884521071029
// MI455X (gfx1250) — compile-verified
//
#include <hip/hip_runtime.h>
#include <hip/hip_bf16.h>
#include <math.h>

#define BB 128
#define DD 512
#define HW 784   // 28*28
#define SPLIT 4  // d-chunks for logits partial sums

typedef float v2f __attribute__((ext_vector_type(2)));
typedef float v8f __attribute__((ext_vector_type(8)));

// ---------------------------------------------------------------------------
// Small fp32 WMMA GEMM body: C[m,n] = sum_k Aeff[m,k] * B(k,n) (+bias)(+Add)
//   Aeff[m,k] = A[m,k]                (A2MODE==0)
//             = A[m,k]*A2[k]          (A2MODE==1, per-k vector)
//             = A[m,k]*A2[m,k]        (A2MODE==2, matrix w/ lda2)
//   B(k,n)    = Bm[n*ldb + k]  if BT  else  Bm[k*ldb + n]
// One wave -> one 16-row stripe x 4 consecutive 16-col tiles (NT=4).
// Register double-buffered; bases stay fixed (global-inferable) and only
// 32-bit element offsets advance -> global_load saddr+voffset form.
// ---------------------------------------------------------------------------
template <int A2MODE, bool BT, bool HASBIAS, bool HASADD>
__device__ __forceinline__ void gemm_body(
    const float* __restrict__ A, int lda,
    const float* __restrict__ A2, int lda2,
    const float* __restrict__ Bm, int ldb,
    const float* __restrict__ bias,
    const float* __restrict__ Add, int ldadd,
    float* __restrict__ C, int ldc,
    int N, int K, int wave)
{
    const int NT = 4;
    int lane    = threadIdx.x & 31;
    int groupsN = N / (16 * NT);
    int tm      = wave / groupsN;
    int tn0     = (wave % groupsN) * NT;
    int lanelo  = lane & 15;
    int half    = lane >> 4;      // 0 or 1
    int khalf   = half * 2;       // A/B vgpr K layout: k = kb + khalf + {0,1}
    int mrow    = tm * 16 + lanelo;

    int aoff  = mrow * lda + khalf;
    int a2off = (A2MODE == 1) ? khalf : (mrow * lda2 + khalf);
    int ncol[NT], boff[NT];
#pragma unroll
    for (int j = 0; j < NT; ++j) {
        ncol[j] = (tn0 + j) * 16 + lanelo;
        boff[j] = BT ? (ncol[j] * ldb + khalf) : (khalf * ldb + ncol[j]);
    }
    const int bstep = BT ? 4 : 4 * ldb;   // offset advance per K-step
    const int boff1 = BT ? 1 : ldb;       // second element of fragment

    // ---- prologue: load K-step 0 fragments -------------------------------
    v2f a_c, b_c[NT];
    a_c.x = A[aoff]; a_c.y = A[aoff + 1];
    if (A2MODE) { a_c.x *= A2[a2off]; a_c.y *= A2[a2off + 1]; }
#pragma unroll
    for (int j = 0; j < NT; ++j) { b_c[j].x = Bm[boff[j]]; b_c[j].y = Bm[boff[j] + boff1]; }
    aoff += 4; a2off += 4;
#pragma unroll
    for (int j = 0; j < NT; ++j) boff[j] += bstep;

    v8f acc[NT] = {};
    for (int kb = 4; kb <= K; kb += 4) {
        v2f a_n, b_n[NT];
        if (kb < K) {                       // prefetch next K-step
            a_n.x = A[aoff]; a_n.y = A[aoff + 1];
            if (A2MODE) { a_n.x *= A2[a2off]; a_n.y *= A2[a2off + 1]; }
#pragma unroll
            for (int j = 0; j < NT; ++j) {
                b_n[j].x = Bm[boff[j]];
                b_n[j].y = Bm[boff[j] + boff1];
            }
        }
#pragma unroll
        for (int j = 0; j < NT; ++j)
            acc[j] = __builtin_amdgcn_wmma_f32_16x16x4_f32(
                false, a_c, false, b_c[j], (short)0, acc[j], false, false);
        if (kb < K) {
            aoff += 4; a2off += 4;
#pragma unroll
            for (int j = 0; j < NT; ++j) boff[j] += bstep;
            a_c = a_n;
#pragma unroll
            for (int j = 0; j < NT; ++j) b_c[j] = b_n[j];
        }
    }

    // ---- epilogue: C layout vgpr i -> row (tm*16 + 8*half + i) -----------
#pragma unroll
    for (int j = 0; j < NT; ++j) {
        int n     = ncol[j];
        int mbase = tm * 16 + 8 * half;
#pragma unroll
        for (int i = 0; i < 8; ++i) {
            int m   = mbase + i;
            float v = acc[j][i];
            if (HASBIAS) v += bias[n];
            if (HASADD)  v += Add[m * ldadd + n];
            C[m * ldc + n] = v;
        }
    }
}

// K1 (fused): blocks [0,16)  -> m_p = m_prev @ W_m^T + b_m        (64 waves)
//             blocks [16,48) -> uv  = (c_i * w_attn) @ W_merge    (128 waves)
__global__ void __launch_bounds__(128) k1_kernel(
    const float* __restrict__ m_prev, const float* __restrict__ W_m,
    const float* __restrict__ b_m,    float* __restrict__ m_p,
    const float* __restrict__ c_i,    const float* __restrict__ w_attn,
    const float* __restrict__ W_merge, float* __restrict__ uv)
{
    int w = threadIdx.x >> 5;
    if (blockIdx.x < 16)
        gemm_body<0, true, true, false>(m_prev, DD, nullptr, 0, W_m, DD,
                                        b_m, nullptr, 0, m_p, DD,
                                        DD, DD, blockIdx.x * 4 + w);
    else
        gemm_body<1, false, false, false>(c_i, DD, w_attn, 0, W_merge, 2 * DD,
                                          nullptr, nullptr, 0, uv, 2 * DD,
                                          2 * DD, DD, (blockIdx.x - 16) * 4 + w);
}

// K2: r = (m_p * u) @ W_kb + v   (u = uv[:,:512], v = uv[:,512:])  (64 waves)
__global__ void __launch_bounds__(128) k2_kernel(
    const float* __restrict__ m_p, const float* __restrict__ uv,
    const float* __restrict__ W_kb, float* __restrict__ r)
{
    gemm_body<2, false, false, true>(m_p, DD, uv, 2 * DD, W_kb, DD,
                                     nullptr, uv + DD, 2 * DD, r, DD,
                                     DD, DD, blockIdx.x * 4 + (threadIdx.x >> 5));
}

// ---------------------------------------------------------------------------
// K3: partial logits.  part[s][b][n] = sum_{d in chunk s} r[b,d] * KB[b,d,n]
// KB native layout (B,D,H,W): row (b,d) is 784 contiguous floats -> coalesced.
// ---------------------------------------------------------------------------
__global__ void logits_partial_kernel(const float* __restrict__ KB,
                                      const float* __restrict__ r,
                                      float* __restrict__ part)
{
    const int DC = DD / SPLIT;          // 128
    int b = blockIdx.x, s = blockIdx.y, t = threadIdx.x;   // 256 threads
    __shared__ float rs[DC];
    if (t < DC) rs[t] = r[b * DD + s * DC + t];
    __syncthreads();

    float a0 = 0.f, a1 = 0.f, a2 = 0.f, a3 = 0.f;
    const float* kb = KB + ((size_t)b * DD + (size_t)s * DC) * HW;
    bool tail = (t < HW - 768);
#pragma unroll 4
    for (int d = 0; d < DC; ++d) {
        float rv = rs[d];
        const float* row = kb + d * HW;
        a0 += rv * row[t];
        a1 += rv * row[t + 256];
        a2 += rv * row[t + 512];
        if (tail) a3 += rv * row[t + 768];
    }
    float* out = part + ((size_t)s * BB + b) * HW;
    out[t] = a0; out[t + 256] = a1; out[t + 512] = a2;
    if (tail) out[t + 768] = a3;
}

// ---------------------------------------------------------------------------
// K4: sum partials, softmax over HW per batch row.  mv[b][n]
// ---------------------------------------------------------------------------
__global__ void softmax_kernel(const float* __restrict__ part,
                               float* __restrict__ mv)
{
    int b = blockIdx.x, t = threadIdx.x;   // 256 threads
    __shared__ float red[256];

    float l[4];
#pragma unroll
    for (int j = 0; j < 4; ++j) {
        int n = t + 256 * j;
        float v = -3.4e38f;
        if (n < HW) {
            v = 0.f;
            for (int s = 0; s < SPLIT; ++s)
                v += part[((size_t)s * BB + b) * HW + n];
        }
        l[j] = v;
    }
    float mx = fmaxf(fmaxf(l[0], l[1]), fmaxf(l[2], l[3]));
    red[t] = mx; __syncthreads();
    for (int off = 128; off > 0; off >>= 1) {
        if (t < off) red[t] = fmaxf(red[t], red[t + off]);
        __syncthreads();
    }
    mx = red[0]; __syncthreads();

    float e[4], sum = 0.f;
#pragma unroll
    for (int j = 0; j < 4; ++j) {
        int n = t + 256 * j;
        e[j] = (n < HW) ? expf(l[j] - mx) : 0.f;
        sum += e[j];
    }
    red[t] = sum; __syncthreads();
    for (int off = 128; off > 0; off >>= 1) {
        if (t < off) red[t] += red[t + off];
        __syncthreads();
    }
    float inv = 1.f / red[0];
#pragma unroll
    for (int j = 0; j < 4; ++j) {
        int n = t + 256 * j;
        if (n < HW) mv[(size_t)b * HW + n] = e[j] * inv;
    }
}

// ---------------------------------------------------------------------------
// K5: m_new[b,d] = sum_n mv[b,n] * KB[b,d,n].  One wave per output row,
// 8 rows per block sharing mv[b] staged in LDS.  __shfl_xor wave32 reduce.
// Traversal is REVERSED (high b / high d first) so the tail of KB that K3
// left resident in the 192 MB L2 is consumed before it gets evicted.
// ---------------------------------------------------------------------------
__global__ void weighted_sum_kernel(const float* __restrict__ KB,
                                    const float* __restrict__ mv,
                                    float* __restrict__ out)
{
    int b = (int)gridDim.x - 1 - (int)blockIdx.x;
    __shared__ float mvs[HW];
    for (int n = threadIdx.x; n < HW; n += 256) mvs[n] = mv[(size_t)b * HW + n];
    __syncthreads();

    int wave = threadIdx.x >> 5, lane = threadIdx.x & 31;
    int d = ((int)gridDim.y - 1 - (int)blockIdx.y) * 8 + wave;
    const float* row = KB + ((size_t)b * DD + d) * HW;
    float acc = 0.f;
    for (int n = lane; n < HW; n += 32) acc += mvs[n] * row[n];
#pragma unroll
    for (int off = 16; off > 0; off >>= 1) acc += __shfl_xor(acc, off, 32);
    if (lane == 0) out[(size_t)b * DD + d] = acc;
}

// ---------------------------------------------------------------------------
extern "C" void kernel_launch(void* const* d_in, const int* in_sizes, int n_in,
                              void* d_out, int out_size, void* d_ws, size_t ws_size,
                              hipStream_t stream)
{
    const float* m_prev  = (const float*)d_in[0];
    const float* KB      = (const float*)d_in[1];
    const float* c_i     = (const float*)d_in[2];
    const float* W_m     = (const float*)d_in[3];
    const float* b_m     = (const float*)d_in[4];
    const float* W_kb    = (const float*)d_in[5];
    const float* W_merge = (const float*)d_in[7];
    const float* w_attn  = (const float*)d_in[9];
    // b_kb / b_merge / b_attn contribute per-batch constants to the logits
    // and cancel inside softmax -> never read.

    float* ws   = (float*)d_ws;
    float* m_p  = ws;                         // 128*512
    float* uv   = ws + 65536;                 // 128*1024  ([u | v])
    float* r    = ws + 196608;                // 128*512
    float* part = ws + 262144;                // 4*128*784
    float* mv   = ws + 663552;                // 128*784
    float* out  = (float*)d_out;              // 128*512

    // K1: m_p and uv in one fused launch (independent GEMMs)
    k1_kernel<<<dim3(48), dim3(128), 0, stream>>>(
        m_prev, W_m, b_m, m_p, c_i, w_attn, W_merge, uv);

    // K2: r = (m_p * u) @ W_kb + v
    k2_kernel<<<dim3(16), dim3(128), 0, stream>>>(m_p, uv, W_kb, r);

    // K3: partial logits, streaming pass 1 over KB (205 MB)
    logits_partial_kernel<<<dim3(BB, SPLIT), dim3(256), 0, stream>>>(KB, r, part);

    // K4: softmax over 784 per batch
    softmax_kernel<<<dim3(BB), dim3(256), 0, stream>>>(part, mv);

    // K5: attention-weighted sum, streaming pass 2 over KB (reverse order)
    weighted_sum_kernel<<<dim3(BB, DD / 8), dim3(256), 0, stream>>>(KB, mv, out);
}